// Bert_30090540876229
// MI455X (gfx1250) — compile-verified
//
#include <hip/hip_runtime.h>
#include <hip/hip_bf16.h>

typedef __attribute__((ext_vector_type(2))) float v2f;
typedef __attribute__((ext_vector_type(8))) float v8f;

#define VOCAB 9
#define DIM 4

// ---------------- kernel 0: zero the histogram bins ----------------
__global__ void zero_cnt_kernel(unsigned int* __restrict__ cnt) {
    if (threadIdx.x < 16) cnt[threadIdx.x] = 0u;
}

// ---------------- kernel 1: vocab histogram of x -------------------
__global__ void hist_kernel(const int* __restrict__ x, int S,
                            unsigned int* __restrict__ cnt) {
    __shared__ unsigned int lc[16];
    if (threadIdx.x < 16) lc[threadIdx.x] = 0u;
    __syncthreads();
    int i = blockIdx.x * blockDim.x + threadIdx.x;
    if (i < S) {
        int v = x[i] & 15;   // values are 0..8; mask keeps LDS in-bounds
        atomicAdd(&lc[v], 1u);
    }
    __syncthreads();
    if (threadIdx.x < VOCAB) atomicAdd(&cnt[threadIdx.x], lc[threadIdx.x]);
}

// ---------------- kernel 2: single-wave collapsed model ------------
// Computes the 9 distinct output rows using two V_WMMA_F32_16X16X4_F32 ops.
__global__ void __launch_bounds__(32)
compute_kernel(const float* __restrict__ emb,    const float* __restrict__ proj_w,
               const float* __restrict__ proj_b, const float* __restrict__ forw_w,
               const float* __restrict__ forw_b, const float* __restrict__ prj_w,
               const float* __restrict__ prj_b,  const unsigned int* __restrict__ cnt,
               float* __restrict__ probs /* 9x9 */) {
    __shared__ float T [16][DIM];   // projected embedding table, zero-padded
    __shared__ float G [16][16];    // T @ T^T
    __shared__ float H2[16][DIM];   // post-forw activations, zero-padded
    __shared__ float L [16][16];    // logits (pre-bias)

    const int lane = threadIdx.x;
    const int n    = lane & 15;     // row (A) / col (B,D) index
    const int kh   = lane >> 4;     // K half: 0 -> K={0,1}, 1 -> K={2,3}

    // table = emb @ proj_w.T + proj_b  (9x4, padded to 16x4 with zeros)
    if (lane < 16) {
        for (int d = 0; d < DIM; ++d) {
            float t = 0.0f;
            if (lane < VOCAB) {
                t = proj_b[d];
                for (int k = 0; k < DIM; ++k)
                    t += emb[lane * DIM + k] * proj_w[d * DIM + k];
            }
            T[lane][d] = t;
        }
    }
    __syncthreads();

    // ---- WMMA #1:  G = T @ T^T  (16x16x4, f32) ----
    // A layout (16x4 f32): lane<16 -> row n, VGPR0=K(2*kh), VGPR1=K(2*kh+1)
    // B = T^T, so B[k][n] = T[n][k]  ==> identical register image to A.
    v2f a;  a.x = T[n][2 * kh];  a.y = T[n][2 * kh + 1];
    v8f c0 = {};
    v8f g = __builtin_amdgcn_wmma_f32_16x16x4_f32(
        /*neg_a=*/false, a, /*neg_b=*/false, a,
        /*c_mod=*/(short)0, c0, /*reuse_a=*/false, /*reuse_b=*/false);
    // D layout: VGPR r holds M = r + 8*kh, lane holds N = n
    for (int r = 0; r < 8; ++r) G[r + 8 * kh][n] = g[r];
    __syncthreads();

    // ---- collapsed attention + relu + forw, one vocab row per lane ----
    if (lane < 16) {
        float h2[DIM] = {0.f, 0.f, 0.f, 0.f};
        if (lane < VOCAB) {
            float mx = -1e30f;
            for (int v = 0; v < VOCAB; ++v) mx = fmaxf(mx, G[lane][v]);
            float den = 0.0f, num[DIM] = {0.f, 0.f, 0.f, 0.f};
            for (int v = 0; v < VOCAB; ++v) {
                float w = (float)cnt[v] * expf(G[lane][v] - mx);
                den += w;
                for (int d = 0; d < DIM; ++d) num[d] += w * T[v][d];
            }
            float inv = 1.0f / den;
            float r1[DIM];
            for (int d = 0; d < DIM; ++d) r1[d] = fmaxf(num[d] * inv, 0.0f);
            for (int d = 0; d < DIM; ++d) {
                float t = forw_b[d];
                for (int k = 0; k < DIM; ++k) t += r1[k] * forw_w[d * DIM + k];
                h2[d] = t;
            }
        }
        for (int d = 0; d < DIM; ++d) H2[lane][d] = h2[d];
    }
    __syncthreads();

    // ---- WMMA #2:  logits = H2 @ prj_w^T  (16x16x4, f32) ----
    v2f a2; a2.x = H2[n][2 * kh]; a2.y = H2[n][2 * kh + 1];
    v2f b2;
    b2.x = (n < VOCAB) ? prj_w[n * DIM + 2 * kh]     : 0.0f;  // B[k][n] = prj_w[n][k]
    b2.y = (n < VOCAB) ? prj_w[n * DIM + 2 * kh + 1] : 0.0f;
    v8f l = __builtin_amdgcn_wmma_f32_16x16x4_f32(
        false, a2, false, b2, (short)0, c0, false, false);
    for (int r = 0; r < 8; ++r) L[r + 8 * kh][n] = l[r];
    __syncthreads();

    // ---- final softmax over VOCAB classes, one row per lane ----
    if (lane < VOCAB) {
        float lg[VOCAB];
        float mx = -1e30f;
        for (int c = 0; c < VOCAB; ++c) {
            lg[c] = L[lane][c] + prj_b[c];
            mx = fmaxf(mx, lg[c]);
        }
        float den = 0.0f;
        for (int c = 0; c < VOCAB; ++c) { lg[c] = expf(lg[c] - mx); den += lg[c]; }
        float inv = 1.0f / den;
        for (int c = 0; c < VOCAB; ++c) probs[lane * VOCAB + c] = lg[c] * inv;
    }
}

// ---------------- kernel 3: scatter the 9 rows to all S positions ----
__global__ void scatter_kernel(const int* __restrict__ x,
                               const float* __restrict__ probs,
                               float* __restrict__ out, int total) {
    int t = blockIdx.x * blockDim.x + threadIdx.x;
    if (t < total) {
        int s = t / VOCAB;
        int c = t - s * VOCAB;
        out[t] = probs[(x[s] & 15) * VOCAB + c];
    }
}

extern "C" void kernel_launch(void* const* d_in, const int* in_sizes, int n_in,
                              void* d_out, int out_size, void* d_ws, size_t ws_size,
                              hipStream_t stream) {
    const int*   x      = (const int*)  d_in[0];
    const float* emb    = (const float*)d_in[1];
    const float* proj_w = (const float*)d_in[2];
    const float* proj_b = (const float*)d_in[3];
    const float* forw_w = (const float*)d_in[4];
    const float* forw_b = (const float*)d_in[5];
    const float* prj_w  = (const float*)d_in[6];
    const float* prj_b  = (const float*)d_in[7];
    float*       out    = (float*)d_out;
    const int    S      = in_sizes[0];

    unsigned int* cnt   = (unsigned int*)d_ws;
    float*        probs = (float*)((char*)d_ws + 64);

    zero_cnt_kernel<<<1, 32, 0, stream>>>(cnt);
    hist_kernel<<<(S + 255) / 256, 256, 0, stream>>>(x, S, cnt);
    compute_kernel<<<1, 32, 0, stream>>>(emb, proj_w, proj_b, forw_w, forw_b,
                                         prj_w, prj_b, cnt, probs);
    const int total = S * VOCAB;
    scatter_kernel<<<(total + 255) / 256, 256, 0, stream>>>(x, probs, out, total);
}